// FeedForward_34952443855443
// MI455X (gfx1250) — compile-verified
//
#include <hip/hip_runtime.h>

typedef float v2f __attribute__((ext_vector_type(2)));
typedef float v4f __attribute__((ext_vector_type(4)));
typedef float v8f __attribute__((ext_vector_type(8)));

#define DIM 1024
#define HID 4096
#define MT 16     // token tile per workgroup
#define KC 128    // hidden-dim chunk
#define XP 1028   // LDS pitch (dwords) for X tile: 1028 % 64 == 4 -> conflict-free b64 A-frag reads
#define HP 132    // LDS pitch (dwords) for activation chunk: 132 % 64 == 4

__launch_bounds__(256)
__global__ void swiglu_moe_fused(const float* __restrict__ x,
                                 const unsigned char* __restrict__ vmask,
                                 const float* __restrict__ gate_w,
                                 const float* __restrict__ up_w,
                                 const float* __restrict__ down_w,
                                 const float* __restrict__ mm_gate_w,
                                 const float* __restrict__ mm_up_w,
                                 const float* __restrict__ mm_down_w,
                                 float* __restrict__ out)
{
    __shared__ float lds_x[MT * XP];  // 16 x 1024 f32 X tile (padded)
    __shared__ float lds_h[MT * HP];  // 16 x 128 f32 activation chunk (padded)

    const int tid  = threadIdx.x;
    const int lane = tid & 31;
    const int wave = tid >> 5;
    const int tok0 = blockIdx.x * MT;

    // ---- stage 0: async global->LDS staging of the X tile (16 x 1024 f32) ----
    // CDNA5 GLOBAL_LOAD_ASYNC_TO_LDS_B128: per-lane 16B memory->LDS DMA, tracked
    // by ASYNCcnt (no VGPR round-trip). LDS byte address = low 32 bits of the
    // generic pointer to the __shared__ array.
    {
        const float* xb = x + (size_t)tok0 * DIM;
        #pragma unroll
        for (int it = 0; it < MT; ++it) {
            unsigned laddr = (unsigned)(uintptr_t)(&lds_x[it * XP + tid * 4]);
            const float* ga = xb + it * DIM + tid * 4;
            asm volatile("global_load_async_to_lds_b128 %0, %1, off"
                         :: "v"(laddr), "v"(ga) : "memory");
        }
        asm volatile("s_wait_asynccnt 0x0" ::: "memory");
    }
    __syncthreads();

    const int nhalf = lane & 15;              // N (or M) index within a 16-wide tile
    const int khalf = (lane >> 4) << 1;       // +0 (lanes 0-15) or +2 (lanes 16-31) in K

    for (int e = 0; e < 2; ++e) {
        const float* gw = e ? mm_gate_w : gate_w;
        const float* uw = e ? mm_up_w   : up_w;
        const float* dw = e ? mm_down_w : down_w;

        v8f acc[8];
        #pragma unroll
        for (int t = 0; t < 8; ++t) acc[t] = (v8f)0.0f;

        for (int c = 0; c < HID / KC; ++c) {
            const int h0 = c * KC;

            // ---- stage 1: gate & up GEMMs for this wave's 16 hidden columns ----
            {
                const int hw = h0 + wave * 16;
                v8f accG = (v8f)0.0f;
                v8f accU = (v8f)0.0f;
                const float* gB = gw + (size_t)(hw + nhalf) * DIM + khalf;
                const float* uB = uw + (size_t)(hw + nhalf) * DIM + khalf;
                const float* aL = &lds_x[nhalf * XP + khalf];

                // Prefetch next chunk's gate/up rows (L2-resident weights -> L0).
                if (c + 1 < HID / KC) {
                    __builtin_prefetch(gB + (size_t)KC * DIM, 0, 0);
                    __builtin_prefetch(uB + (size_t)KC * DIM, 0, 0);
                }

                #pragma unroll 4
                for (int k = 0; k < DIM; k += 4) {
                    v2f a  = *(const v2f*)(aL + k);
                    v2f bg = *(const v2f*)(gB + k);
                    v2f bu = *(const v2f*)(uB + k);
                    accG = __builtin_amdgcn_wmma_f32_16x16x4_f32(
                        false, a, false, bg, (short)0, accG, false, false);
                    accU = __builtin_amdgcn_wmma_f32_16x16x4_f32(
                        false, a, false, bu, (short)0, accU, false, false);
                }
                // SiLU(gate) * up  ->  LDS activation chunk.
                // Denominator 1+exp(-g) >= 1, so a single v_rcp_f32 is safe and
                // avoids the IEEE div expansion.
                #pragma unroll
                for (int i = 0; i < 8; ++i) {
                    float gv = accG[i];
                    float uv = accU[i];
                    float s  = gv * __builtin_amdgcn_rcpf(1.0f + __expf(-gv)) * uv;
                    int row  = i + ((lane >> 4) << 3);
                    lds_h[row * HP + wave * 16 + nhalf] = s;
                }
            }
            __syncthreads();

            // ---- stage 2: accumulate down-projection for this wave's 128 out cols ----
            {
                const int d0 = wave * 128;
                #pragma unroll
                for (int sub = 0; sub < 2; ++sub) {
                    v2f afrag[16];
                    const float* hL = &lds_h[nhalf * HP + sub * 64 + khalf];
                    #pragma unroll
                    for (int kk = 0; kk < 16; ++kk)
                        afrag[kk] = *(const v2f*)(hL + kk * 4);
                    #pragma unroll
                    for (int t = 0; t < 8; ++t) {
                        const float* dB = dw + (size_t)(d0 + t * 16 + nhalf) * HID
                                             + h0 + sub * 64 + khalf;
                        if (sub == 0 && c + 1 < HID / KC)
                            __builtin_prefetch(dB + KC, 0, 0);  // next chunk's down rows
                        #pragma unroll
                        for (int kk = 0; kk < 16; ++kk) {
                            v2f b = *(const v2f*)(dB + kk * 4);
                            acc[t] = __builtin_amdgcn_wmma_f32_16x16x4_f32(
                                false, afrag[kk], false, b, (short)0, acc[t], false, false);
                        }
                    }
                }
            }
            __syncthreads();
        }

        // ---- store rows whose mask selects this expert ----
        // Output is write-once / never re-read: use non-temporal stores (TH_NT)
        // so the 64 MB of results does not evict the 96 MB of L2-resident
        // weights that every workgroup re-reads.
        {
            const int d0 = wave * 128;
            #pragma unroll
            for (int i = 0; i < 8; ++i) {
                int row = i + ((lane >> 4) << 3);
                int tok = tok0 + row;
                bool isvis = vmask[tok] != 0;
                if (isvis == (e != 0)) {
                    float* orow = out + (size_t)tok * DIM + d0 + nhalf;
                    #pragma unroll
                    for (int t = 0; t < 8; ++t)
                        __builtin_nontemporal_store(acc[t][i], orow + t * 16);
                }
            }
        }
    }
}

extern "C" void kernel_launch(void* const* d_in, const int* in_sizes, int n_in,
                              void* d_out, int out_size, void* d_ws, size_t ws_size,
                              hipStream_t stream) {
    const float* x          = (const float*)d_in[0];
    const unsigned char* vm = (const unsigned char*)d_in[1];
    const float* gate_w     = (const float*)d_in[2];
    const float* up_w       = (const float*)d_in[3];
    const float* down_w     = (const float*)d_in[4];
    const float* mm_gate_w  = (const float*)d_in[5];
    const float* mm_up_w    = (const float*)d_in[6];
    const float* mm_down_w  = (const float*)d_in[7];
    float* out              = (float*)d_out;

    const int ntok = 2 * 8192;                 // BS * SEQ
    dim3 grid(ntok / MT), block(256);
    swiglu_moe_fused<<<grid, block, 0, stream>>>(
        x, vm, gate_w, up_w, down_w, mm_gate_w, mm_up_w, mm_down_w, out);
}